// GNNCritic_19610820673795
// MI455X (gfx1250) — compile-verified
//
#include <hip/hip_runtime.h>
#include <hip/hip_bf16.h>

typedef float v2f __attribute__((ext_vector_type(2)));
typedef float v8f __attribute__((ext_vector_type(8)));

#define CH 32
#define SLOPE 0.01f

// Hardware FP32 atomic add (GLOBAL_ATOMIC_ADD_F32) — targets are coarse-grained
// device workspace, so the "unsafe" (non-CAS) path is correct and fast.
__device__ __forceinline__ void atomic_add_f32(float* p, float v) {
  unsafeAtomicAdd(p, v);
}

// ---------------- degree / norm ----------------
__global__ void k_edge_deg(const int* __restrict__ col, const float* __restrict__ w,
                           float* __restrict__ deg, int E) {
  int e = blockIdx.x * blockDim.x + threadIdx.x;
  if (e < E) atomic_add_f32(&deg[col[e]], w[e]);
}

__global__ void k_dinv(const float* __restrict__ deg, float* __restrict__ dinv, int N) {
  int n = blockIdx.x * blockDim.x + threadIdx.x;
  if (n < N) {
    float d = deg[n];
    dinv[n] = (d > 0.f) ? rsqrtf(fmaxf(d, 1e-12f)) : 0.f;
  }
}

__global__ void k_norm(const int* __restrict__ row, const int* __restrict__ col,
                       const float* __restrict__ w, const float* __restrict__ dinv,
                       float* __restrict__ norm, int E) {
  int e = blockIdx.x * blockDim.x + threadIdx.x;
  if (e < E) norm[e] = dinv[row[e]] * w[e] * dinv[col[e]];
}

// ---------------- read-in MLP (tiny: VALU) ----------------
__global__ void k_readin(const float* __restrict__ state, const float* __restrict__ action,
                         const float* __restrict__ W_in, const float* __restrict__ b_in,
                         float* __restrict__ h, int N) {
  int gid = blockIdx.x * blockDim.x + threadIdx.x;
  int n = gid >> 5, c = gid & 31;
  if (n >= N) return;
  float acc = b_in[c];
#pragma unroll
  for (int i = 0; i < 8; ++i) acc += state[n * 8 + i] * W_in[i * CH + c];
#pragma unroll
  for (int i = 0; i < 2; ++i) acc += action[n * 2 + i] * W_in[(8 + i) * CH + c];
  h[gid] = (acc > 0.f) ? acc : SLOPE * acc;
}

// ---------------- propagate: wave = edge, lane = channel ----------------
// Gather z[row[e]] (one coalesced 128B line per wave), scale by norm[e],
// hardware-atomic scatter-add into zout[col[e]] (one line per wave).
__global__ void k_propagate(const float* __restrict__ z, const int* __restrict__ row,
                            const int* __restrict__ col, const float* __restrict__ norm,
                            float* __restrict__ zout, int E) {
  long long gid = (long long)blockIdx.x * blockDim.x + threadIdx.x;
  int e = (int)(gid >> 5);
  int c = (int)(gid & 31);
  if (e >= E) return;
  float v = z[(long long)row[e] * CH + c] * norm[e];
  atomic_add_f32(&zout[(long long)col[e] * CH + c], v);
}

// ---------------- [N,32] @ [32,32] accumulate via V_WMMA_F32_16X16X4_F32 ----------------
// One wave computes a 16x32 output tile. A-layout (16x4 f32): lane<16 holds K=0,1;
// lane>=16 holds K=2,3 (contiguous float2 per lane). B/C/D per ISA striped layouts.
__global__ void k_tap_matmul(const float* __restrict__ Z, const float* __restrict__ W,
                             float* __restrict__ Out, int N) {
  int wave = (blockIdx.x * blockDim.x + threadIdx.x) >> 5;
  int lane = threadIdx.x & 31;
  int ntiles = (N + 15) >> 4;
  if (wave >= ntiles) return;          // wave-uniform: EXEC stays all-1 for WMMA
  int m0 = wave << 4;
  int mlo = lane & 15;
  int khalf = lane >> 4;               // 0 or 1

  v8f acc0 = {}, acc1 = {};
#pragma unroll
  for (int r = 0; r < 8; ++r) {
    int orow = m0 + r + 8 * khalf;
    int crow = (orow < N) ? orow : (N - 1);
    acc0[r] = Out[crow * CH + mlo];
    acc1[r] = Out[crow * CH + 16 + mlo];
  }

  int arow = (m0 + mlo < N) ? (m0 + mlo) : (N - 1);
#pragma unroll
  for (int k0 = 0; k0 < CH; k0 += 4) {
    v2f a = *reinterpret_cast<const v2f*>(&Z[arow * CH + k0 + 2 * khalf]);
    v2f b0, b1;
    b0.x = W[(k0 + 2 * khalf) * CH + mlo];
    b0.y = W[(k0 + 2 * khalf + 1) * CH + mlo];
    b1.x = W[(k0 + 2 * khalf) * CH + 16 + mlo];
    b1.y = W[(k0 + 2 * khalf + 1) * CH + 16 + mlo];
    acc0 = __builtin_amdgcn_wmma_f32_16x16x4_f32(false, a, false, b0, (short)0, acc0, false, false);
    acc1 = __builtin_amdgcn_wmma_f32_16x16x4_f32(false, a, false, b1, (short)0, acc1, false, false);
  }

#pragma unroll
  for (int r = 0; r < 8; ++r) {
    int orow = m0 + r + 8 * khalf;
    if (orow < N) {
      Out[orow * CH + mlo] = acc0[r];
      Out[orow * CH + 16 + mlo] = acc1[r];
    }
  }
}

// ---------------- bias + leaky relu ----------------
__global__ void k_bias_lrelu(const float* __restrict__ outv, const float* __restrict__ b,
                             float* __restrict__ h, int N) {
  int gid = blockIdx.x * blockDim.x + threadIdx.x;
  int n = gid >> 5, c = gid & 31;
  if (n >= N) return;
  float v = outv[gid] + b[c];
  h[gid] = (v > 0.f) ? v : SLOPE * v;
}

// ---------------- readout: y = h@W_out + b_out, scatter-mean over batch ----------------
__global__ void k_readout(const float* __restrict__ h, const float* __restrict__ W_out,
                          const float* __restrict__ b_out, const int* __restrict__ batch,
                          float* __restrict__ sums, float* __restrict__ cnt, int N) {
  int gid = blockIdx.x * blockDim.x + threadIdx.x;
  int n = gid >> 5, c = gid & 31;
  if (n >= N) return;
  float v = h[gid] * W_out[c];
#pragma unroll
  for (int off = 16; off > 0; off >>= 1) v += __shfl_xor(v, off, 32);
  if (c == 0) {
    int b = batch[n];
    atomic_add_f32(&sums[b], v + b_out[0]);
    atomic_add_f32(&cnt[b], 1.0f);
  }
}

__global__ void k_finalize(const float* __restrict__ sums, const float* __restrict__ cnt,
                           float* __restrict__ outp, int B) {
  int b = blockIdx.x * blockDim.x + threadIdx.x;
  if (b < B) outp[b] = sums[b] / fmaxf(cnt[b], 1.0f);
}

extern "C" void kernel_launch(void* const* d_in, const int* in_sizes, int n_in,
                              void* d_out, int out_size, void* d_ws, size_t ws_size,
                              hipStream_t stream) {
  const float* state  = (const float*)d_in[0];
  const float* action = (const float*)d_in[1];
  const int*   eidx   = (const int*)d_in[2];
  const float* eattr  = (const float*)d_in[3];
  const int*   batch  = (const int*)d_in[4];
  const float* W_in   = (const float*)d_in[5];
  const float* b_in   = (const float*)d_in[6];
  const float* W_taps = (const float*)d_in[7];
  const float* b_gnn  = (const float*)d_in[8];
  const float* W_out  = (const float*)d_in[9];
  const float* b_out  = (const float*)d_in[10];

  const int N = in_sizes[0] / 8;   // state is [N, 8]
  const int E = in_sizes[3];       // edge_attr is [E]
  const int B = out_size;
  const int* row = eidx;
  const int* col = eidx + E;

  // --- carve workspace (256B aligned slabs) ---
  char* ws = (char*)d_ws;
  auto carve = [&](size_t bytes) -> float* {
    char* p = ws;
    ws += (bytes + 255) & ~(size_t)255;
    return (float*)p;
  };
  float* deg  = carve((size_t)N * 4);
  float* dinv = carve((size_t)N * 4);
  float* nrm  = carve((size_t)E * 4);
  float* h    = carve((size_t)N * CH * 4);
  float* outb = carve((size_t)N * CH * 4);
  float* z0   = carve((size_t)N * CH * 4);
  float* z1   = carve((size_t)N * CH * 4);
  float* sums = carve((size_t)B * 4);
  float* cnt  = carve((size_t)B * 4);

  const int T = 256;
  const int gE   = (E + T - 1) / T;
  const int gN   = (N + T - 1) / T;
  const int gNC  = (int)(((long long)N * CH + T - 1) / T);
  const int gEC  = (int)(((long long)E * CH + T - 1) / T);
  const int tiles = (N + 15) / 16;
  const int gMM  = (tiles + 7) / 8;   // 8 waves (256 threads) per block

  // gcn_norm
  hipMemsetAsync(deg, 0, (size_t)N * 4, stream);
  k_edge_deg<<<gE, T, 0, stream>>>(col, eattr, deg, E);
  k_dinv<<<gN, T, 0, stream>>>(deg, dinv, N);
  k_norm<<<gE, T, 0, stream>>>(row, col, eattr, dinv, nrm, E);

  // read-in MLP
  k_readin<<<gNC, T, 0, stream>>>(state, action, W_in, b_in, h, N);

  // TAGConv layers
  for (int l = 0; l < 2; ++l) {
    const float* Wl = W_taps + (size_t)l * 5 * CH * CH;
    hipMemsetAsync(outb, 0, (size_t)N * CH * 4, stream);
    k_tap_matmul<<<gMM, T, 0, stream>>>(h, Wl, outb, N);           // tap 0
    const float* zsrc = h;
    for (int k = 1; k <= 4; ++k) {
      float* zdst = (k & 1) ? z0 : z1;
      hipMemsetAsync(zdst, 0, (size_t)N * CH * 4, stream);
      k_propagate<<<gEC, T, 0, stream>>>(zsrc, row, col, nrm, zdst, E);
      k_tap_matmul<<<gMM, T, 0, stream>>>(zdst, Wl + (size_t)k * CH * CH, outb, N);
      zsrc = zdst;
    }
    k_bias_lrelu<<<gNC, T, 0, stream>>>(outb, b_gnn + l * CH, h, N);
  }

  // readout + scatter-mean
  hipMemsetAsync(sums, 0, (size_t)B * 4, stream);
  hipMemsetAsync(cnt, 0, (size_t)B * 4, stream);
  k_readout<<<gNC, T, 0, stream>>>(h, W_out, b_out, batch, sums, cnt, N);
  k_finalize<<<(B + T - 1) / T, T, 0, stream>>>(sums, cnt, (float*)d_out, B);
}